// Classifier_1451698946469
// MI455X (gfx1250) — compile-verified
//
#include <hip/hip_runtime.h>

typedef float    v2f __attribute__((ext_vector_type(2)));
typedef float    v8f __attribute__((ext_vector_type(8)));
typedef unsigned v4u __attribute__((ext_vector_type(4)));
typedef unsigned v8u __attribute__((ext_vector_type(8)));

#define BDIM   128      // batch (M = N = 128)
#define TILE_K 128      // K per LDS stage
#define LSTR   132      // padded LDS row stride (floats): bank = (4*row + k) % 64, conflict-free
#define NWG    512      // split-K workgroups

__device__ __forceinline__ void gadd_f32(float* p, float v) {
    __hip_atomic_fetch_add(p, v, __ATOMIC_RELAXED, __HIP_MEMORY_SCOPE_AGENT);
}

// TDM: one descriptor moves a 16-row x 128-col f32 tile global->LDS with HW padding
// (128 DWORDs data + 4 DWORDs pad per row == LSTR=132 floats). D# per CDNA5 ISA ch8.
__device__ __forceinline__ void tdm_load_tile(unsigned lds_addr, const float* gaddr,
                                              unsigned d_elems) {
    const unsigned long long ga = (unsigned long long)(uintptr_t)gaddr;
    const v4u g0 = { 1u,                                   // count=1, user mode
                     lds_addr,                             // LDS byte address
                     (unsigned)ga,                         // global_addr[31:0]
                     (unsigned)(ga >> 32) | (2u << 30) };  // global_addr[56:32] | type=2
    const v8u g1 = {
        (2u << 16) | (1u << 20) | (6u << 22) | (3u << 25), // data_size=4B, pad_en, every 128 DW pad 4 DW
        (d_elems & 0xFFFFu) << 16,                         // tensor_dim0[15:0]  (bits 63:48)
        (d_elems >> 16) | (128u << 16),                    // tensor_dim0[31:16] | tensor_dim1=128
        128u << 16,                                        // tile_dim0 = 128 (bits 127:112)
        16u,                                               // tile_dim1 = 16 rows
        d_elems,                                           // tensor_dim0_stride[31:0] = D
        0u, 0u };                                          // stride hi / dim1_stride = 0
    asm volatile("tensor_load_to_lds %0, %1" :: "s"(g0), "s"(g1) : "memory");
}

__global__ __launch_bounds__(256) void zero_ws_kernel(float* p, int n) {
    int i = blockIdx.x * blockDim.x + threadIdx.x;
    if (i < n) p[i] = 0.0f;
}

__global__ __launch_bounds__(256) void gemm_partial_kernel(
    const float* __restrict__ X, const float* __restrict__ Y,
    float* __restrict__ dots, float* __restrict__ xsq, float* __restrict__ ysq,
    int D)
{
    __shared__ float Xs[BDIM * LSTR];
    __shared__ float Ys[BDIM * LSTR];
    __shared__ float lxsq[BDIM];
    __shared__ float lysq[BDIM];

    const int tid    = threadIdx.x;
    const int wave   = tid >> 5;
    const int lane   = tid & 31;
    const int lane15 = lane & 15;
    const int hi     = lane >> 4;        // K sub-pair: covers K % 4 in {2hi, 2hi+1}
    const int wr     = wave >> 1;        // 4 row groups: rows [32*wr, 32*wr+32)
    const int wc     = wave & 1;         // 2 col groups: cols [64*wc, 64*wc+64)

    if (tid < BDIM) { lxsq[tid] = 0.0f; lysq[tid] = 0.0f; }
    __syncthreads();

    v8f acc[2][4];
    #pragma unroll
    for (int m = 0; m < 2; ++m)
        #pragma unroll
        for (int n = 0; n < 4; ++n)
            #pragma unroll
            for (int e = 0; e < 8; ++e) acc[m][n][e] = 0.0f;

    // sums of squares harvested from the WMMA fragments; v2f accumulators so each
    // update is a single v_pk_fma_f32 on the ds_load_b64 pair (no repacking movs)
    v2f nsx0 = {0.0f, 0.0f}, nsx1 = {0.0f, 0.0f};
    v2f nsy[4] = {{0.0f,0.0f},{0.0f,0.0f},{0.0f,0.0f},{0.0f,0.0f}};

    // each wave stages its own 16-row slice of both tiles via TDM (wave-uniform D#)
    const unsigned ldsRowOff = (unsigned)(wave * 16 * LSTR * 4);
    const unsigned xlds = (unsigned)(uintptr_t)&Xs[0] + ldsRowOff;
    const unsigned ylds = (unsigned)(uintptr_t)&Ys[0] + ldsRowOff;
    const size_t   growOff = (size_t)(wave * 16) * (size_t)D;

    const int nchunks = D / TILE_K;
    for (int c = blockIdx.x; c < nchunks; c += gridDim.x) {
        const size_t k0 = (size_t)c * TILE_K;

        tdm_load_tile(xlds, X + growOff + k0, (unsigned)D);
        tdm_load_tile(ylds, Y + growOff + k0, (unsigned)D);
        __builtin_amdgcn_s_wait_tensorcnt(0);
        __syncthreads();

        const int arow0 = (wr << 5) + lane15;          // tile m=0 row
        const int bcol0 = (wc << 6) + lane15;          // tile n=0 col
        const int ksub  = hi << 1;
        for (int k = 0; k < TILE_K; k += 4) {
            const v2f a0 = *(const v2f*)&Xs[arow0 * LSTR + k + ksub];
            const v2f a1 = *(const v2f*)&Xs[(arow0 + 16) * LSTR + k + ksub];
            nsx0 += a0 * a0;
            nsx1 += a1 * a1;
            #pragma unroll
            for (int n = 0; n < 4; ++n) {
                const v2f b = *(const v2f*)&Ys[(bcol0 + (n << 4)) * LSTR + k + ksub];
                nsy[n] += b * b;
                acc[0][n] = __builtin_amdgcn_wmma_f32_16x16x4_f32(
                    false, a0, false, b, (short)0, acc[0][n], false, false);
                acc[1][n] = __builtin_amdgcn_wmma_f32_16x16x4_f32(
                    false, a1, false, b, (short)0, acc[1][n], false, false);
            }
        }
        __syncthreads();
    }

    // Scatter partial 128x128 tile. D layout: VGPR r holds M=r (lanes 0-15), M=r+8 (lanes 16-31).
    #pragma unroll
    for (int m = 0; m < 2; ++m) {
        const int rbase = (wr << 5) + (m << 4) + (hi << 3);
        #pragma unroll
        for (int n = 0; n < 4; ++n) {
            const int col = (wc << 6) + (n << 4) + lane15;
            #pragma unroll
            for (int r = 0; r < 8; ++r)
                gadd_f32(&dots[(rbase + r) * BDIM + col], acc[m][n][r]);
        }
    }

    // Norm reduction. hi=0/1 lanes cover complementary K halves; duplicate waves give
    // exact power-of-two factors: A rows read by 2 waves (wc=0,1), B cols by 4 (wr=0..3).
    atomicAdd(&lxsq[(wr << 5) + lane15],      (nsx0.x + nsx0.y) * 0.5f);
    atomicAdd(&lxsq[(wr << 5) + 16 + lane15], (nsx1.x + nsx1.y) * 0.5f);
    #pragma unroll
    for (int n = 0; n < 4; ++n)
        atomicAdd(&lysq[(wc << 6) + (n << 4) + lane15], (nsy[n].x + nsy[n].y) * 0.25f);
    __syncthreads();

    if (tid < BDIM) {
        gadd_f32(&xsq[tid], lxsq[tid]);
        gadd_f32(&ysq[tid], lysq[tid]);
    }
}

__global__ __launch_bounds__(128) void finalize_kernel(
    const float* __restrict__ dots, const float* __restrict__ xsq,
    const float* __restrict__ ysq, float* __restrict__ out)
{
    __shared__ float xn[BDIM];
    __shared__ float yn[BDIM];
    __shared__ int cnt1, cnt10;

    const int i = threadIdx.x;
    xn[i] = sqrtf(xsq[i]);
    yn[i] = sqrtf(ysq[i]);
    if (i == 0) { cnt1 = 0; cnt10 = 0; }
    __syncthreads();

    // sim.T[i][j] = dots[j][i] / max(xn[j]*yn[i], eps); thread i owns row i of sim.T
    const float yni = yn[i];
    const float sd  = dots[i * BDIM + i] / fmaxf(xn[i] * yni, 1e-8f);
    int rank = 0;
    for (int j = 0; j < BDIM; ++j) {
        const float s = dots[j * BDIM + i] / fmaxf(xn[j] * yni, 1e-8f);
        // first-index tie-break identical to argmax / lax.top_k
        if (s > sd || (s == sd && j < i)) ++rank;
    }
    if (rank == 0)  atomicAdd(&cnt1, 1);
    if (rank < 10)  atomicAdd(&cnt10, 1);
    __syncthreads();
    if (i == 0) {
        out[0] = (float)cnt1  / 128.0f;   // top1 accuracy
        out[1] = (float)cnt10 / 128.0f;   // top10 accuracy
    }
}

extern "C" void kernel_launch(void* const* d_in, const int* in_sizes, int n_in,
                              void* d_out, int out_size, void* d_ws, size_t ws_size,
                              hipStream_t stream) {
    const float* Z = (const float*)d_in[0];
    const float* Y = (const float*)d_in[1];
    float* out  = (float*)d_out;
    float* dots = (float*)d_ws;                  // 128*128 f32
    float* xsq  = dots + BDIM * BDIM;            // 128 f32
    float* ysq  = xsq + BDIM;                    // 128 f32

    const int D = in_sizes[0] / BDIM;            // 262144

    const int nzero = BDIM * BDIM + 2 * BDIM;
    zero_ws_kernel<<<(nzero + 255) / 256, 256, 0, stream>>>(dots, nzero);
    gemm_partial_kernel<<<NWG, 256, 0, stream>>>(Z, Y, dots, xsq, ysq, D);
    finalize_kernel<<<1, 128, 0, stream>>>(dots, xsq, ysq, out);
}